// RotaryCacheMHA_89541478187538
// MI455X (gfx1250) — compile-verified
//
#include <hip/hip_runtime.h>
#include <hip/hip_bf16.h>

#define B_    16
#define TMAX  4096
#define DM    1024
#define H_    16
#define DH_   64
#define HALF  32
#define NPART 8          // flash-decode partitions per (b,h)

typedef __attribute__((ext_vector_type(2))) float v2f;
typedef __attribute__((ext_vector_type(8))) float v8f;

// ln(10000)/32
#define LOG1E4_OVER_HALF 0.28782313662425575f

// ---------------------------------------------------------------------------
// Kernel 1: QKV projections  P = x(16x1024) @ W(1024x1024) + bias
// One wave32 per 16x16 output tile; K-loop of 256 x V_WMMA_F32_16X16X4_F32.
// A (16x4 f32): lanes 0-15 row M, v0=K0 v1=K1 ; lanes 16-31 v0=K2 v1=K3.
// B (4x16 f32): lanes 0-15 col N, v0=K0 v1=K1 ; lanes 16-31 v0=K2 v1=K3.
// C/D (16x16 f32): VGPR r -> (M=r, N=lane) / (M=r+8, N=lane-16).
// ---------------------------------------------------------------------------
__global__ __launch_bounds__(32) void qkv_proj_wmma(
    const float* __restrict__ x,
    const float* __restrict__ Wq, const float* __restrict__ bq,
    const float* __restrict__ Wk, const float* __restrict__ bk,
    const float* __restrict__ Wv, const float* __restrict__ bv,
    float* __restrict__ ws_q, float* __restrict__ ws_k, float* __restrict__ ws_v)
{
    const int tile = blockIdx.x;          // 0..63 output col tile
    const int sel  = blockIdx.y;          // 0=q 1=k 2=v
    const float* W; const float* bias; float* out;
    if (sel == 0)      { W = Wq; bias = bq; out = ws_q; }
    else if (sel == 1) { W = Wk; bias = bk; out = ws_k; }
    else               { W = Wv; bias = bv; out = ws_v; }

    const int lane = threadIdx.x;
    const int ml   = lane & 15;
    const int kh   = lane >> 4;
    const int col  = tile * 16 + ml;

    v8f c = {};
    for (int kk = 0; kk < DM; kk += 4) {
        const int kb = kk + 2 * kh;
        v2f a, b;
        a.x = x[ml * DM + kb];
        a.y = x[ml * DM + kb + 1];
        b.x = W[(size_t)kb * DM + col];
        b.y = W[(size_t)(kb + 1) * DM + col];
        c = __builtin_amdgcn_wmma_f32_16x16x4_f32(false, a, false, b,
                                                  (short)0, c, false, false);
    }
    const float bias_c = bias[col];
#pragma unroll
    for (int r = 0; r < 8; ++r)
        out[(r + 8 * kh) * DM + col] = c[r] + bias_c;
}

// ---------------------------------------------------------------------------
// Kernel 2: cos/sin table for all t (L2-resident, 1MB) + RoPE(q, pos=cache_index)
// ---------------------------------------------------------------------------
__global__ __launch_bounds__(256) void rope_setup(
    float* __restrict__ cos_tab, float* __restrict__ sin_tab,
    float* __restrict__ ws_q, const int* __restrict__ cache_index)
{
    const int gid = blockIdx.x * 256 + threadIdx.x;
    const int table_elems = TMAX * HALF;          // 131072
    if (gid < table_elems) {
        const int t = gid >> 5;
        const int j = gid & 31;
        const float invf = expf(-LOG1E4_OVER_HALF * (float)j);
        float s, cx;
        sincosf((float)t * invf, &s, &cx);
        cos_tab[gid] = cx;
        sin_tab[gid] = s;
    } else {
        const int idx = gid - table_elems;        // 0..8191: (b,h,j) pairs of q
        if (idx < B_ * H_ * HALF) {
            const int b = idx >> 9;
            const int h = (idx >> 5) & 15;
            const int j = idx & 31;
            const int pos = *cache_index;
            const float invf = expf(-LOG1E4_OVER_HALF * (float)j);
            float s, cx;
            sincosf((float)pos * invf, &s, &cx);
            float* qp = ws_q + (b * H_ + h) * DH_;
            const float q1 = qp[j], q2 = qp[j + HALF];
            qp[j]        = q1 * cx - q2 * s;
            qp[j + HALF] = q1 * s  + q2 * cx;
        }
    }
}

// ---------------------------------------------------------------------------
// Kernel 3: flash-decode partial attention. Block = (b,h,partition).
// Each thread owns t = p + NPART*li -> 256B contiguous K-row reads (b128).
// RoPE applied on the fly from the L2-resident table. Partition emits
// unnormalized (o[64], max, sum) to scratch.
// ---------------------------------------------------------------------------
__global__ __launch_bounds__(256) void attn_partial(
    const float* __restrict__ cached_key, const float* __restrict__ cached_value,
    const float* __restrict__ ws_q, const float* __restrict__ ws_k,
    const float* __restrict__ ws_v,
    const float* __restrict__ cos_tab, const float* __restrict__ sin_tab,
    float* __restrict__ ws_part, const int* __restrict__ cache_index)
{
    __shared__ float q_s[DH_];
    __shared__ float sc[TMAX / NPART];            // 512 local scores
    __shared__ float red[256];

    const int tid = threadIdx.x;
    const int p   = blockIdx.x & (NPART - 1);
    const int bh  = blockIdx.x >> 3;
    const int b   = bh >> 4;
    const int h   = bh & 15;
    const int ci  = *cache_index;
    const int T   = ci + 1;
    const int L   = (T > p) ? ((T - p + NPART - 1) / NPART) : 0;

    if (tid < DH_) q_s[tid] = ws_q[(b * H_ + h) * DH_ + tid];
    __syncthreads();

    // scores for this partition's t values
    for (int li = tid; li < L; li += 256) {
        const int t = p + NPART * li;
        const float* kp = (t == ci)
            ? (ws_k + (b * H_ + h) * DH_)
            : (cached_key + (((size_t)b * TMAX + t) * H_ + h) * DH_);
        const float4* kp4 = (const float4*)kp;
        const float4* ct4 = (const float4*)(cos_tab + t * HALF);
        const float4* st4 = (const float4*)(sin_tab + t * HALF);
        const float4* q4  = (const float4*)q_s;
        float s = 0.f;
#pragma unroll
        for (int jj = 0; jj < 8; ++jj) {
            const float4 k1 = kp4[jj];
            const float4 k2 = kp4[jj + 8];
            const float4 cc = ct4[jj];
            const float4 ss = st4[jj];
            const float4 q1 = q4[jj];
            const float4 q2 = q4[jj + 8];
            s += q1.x * (k1.x * cc.x - k2.x * ss.x) + q2.x * (k1.x * ss.x + k2.x * cc.x);
            s += q1.y * (k1.y * cc.y - k2.y * ss.y) + q2.y * (k1.y * ss.y + k2.y * cc.y);
            s += q1.z * (k1.z * cc.z - k2.z * ss.z) + q2.z * (k1.z * ss.z + k2.z * cc.z);
            s += q1.w * (k1.w * cc.w - k2.w * ss.w) + q2.w * (k1.w * ss.w + k2.w * cc.w);
        }
        sc[li] = s * 0.125f;                      // 1/sqrt(64)
    }
    __syncthreads();

    // partition max
    float m = -3.0e38f;
    for (int li = tid; li < L; li += 256) m = fmaxf(m, sc[li]);
    red[tid] = m; __syncthreads();
    for (int off = 128; off > 0; off >>= 1) {
        if (tid < off) red[tid] = fmaxf(red[tid], red[tid + off]);
        __syncthreads();
    }
    const float Mp = red[0];
    __syncthreads();

    // exp + partition sum (unnormalized)
    float sum = 0.f;
    for (int li = tid; li < L; li += 256) {
        const float pr = __expf(sc[li] - Mp);
        sc[li] = pr;
        sum += pr;
    }
    red[tid] = sum; __syncthreads();
    for (int off = 128; off > 0; off >>= 1) {
        if (tid < off) red[tid] += red[tid + off];
        __syncthreads();
    }
    const float Sp = red[0];
    __syncthreads();

    // unnormalized o_p[d] = sum_t p_t * v[t,d]; threads = 4 t-groups x 64 d
    const int d  = tid & 63;
    const int tg = tid >> 6;
    float acc = 0.f;
    for (int li = tg; li < L; li += 4) {
        const int t = p + NPART * li;
        const float* vp = (t == ci)
            ? (ws_v + (b * H_ + h) * DH_)
            : (cached_value + (((size_t)b * TMAX + t) * H_ + h) * DH_);
        acc += sc[li] * vp[d];
    }
    red[tid] = acc; __syncthreads();
    if (tid < 64) {
        float* dst = ws_part + ((size_t)(bh * NPART + p)) * (DH_ + 2);
        dst[tid] = red[tid] + red[tid + 64] + red[tid + 128] + red[tid + 192];
        if (tid == 0) { dst[DH_] = Mp; dst[DH_ + 1] = Sp; }
    }
}

// ---------------------------------------------------------------------------
// Kernel 4: combine the NPART partials per (b,h) (log-sum-exp merge)
// ---------------------------------------------------------------------------
__global__ __launch_bounds__(64) void attn_combine(
    const float* __restrict__ ws_part, float* __restrict__ ws_o)
{
    const int bh = blockIdx.x;
    const int d  = threadIdx.x;
    float Mp[NPART], Sp[NPART];
    float M = -3.0e38f;
#pragma unroll
    for (int p = 0; p < NPART; ++p) {
        const float* src = ws_part + ((size_t)(bh * NPART + p)) * (DH_ + 2);
        Mp[p] = src[DH_];
        Sp[p] = src[DH_ + 1];
        M = fmaxf(M, Mp[p]);
    }
    float S = 0.f, o = 0.f;
#pragma unroll
    for (int p = 0; p < NPART; ++p) {
        const float w = __expf(Mp[p] - M);
        S += Sp[p] * w;
        o += ws_part[((size_t)(bh * NPART + p)) * (DH_ + 2) + d] * w;
    }
    ws_o[bh * DH_ + d] = o / S;
}

// ---------------------------------------------------------------------------
// Kernel 5: out = o(16x1024) @ Wo(1024x1024) + bo, same WMMA f32 structure
// ---------------------------------------------------------------------------
__global__ __launch_bounds__(32) void out_proj_wmma(
    const float* __restrict__ o, const float* __restrict__ Wo,
    const float* __restrict__ bo, float* __restrict__ out)
{
    const int tile = blockIdx.x;
    const int lane = threadIdx.x;
    const int ml = lane & 15;
    const int kh = lane >> 4;
    const int col = tile * 16 + ml;

    v8f c = {};
    for (int kk = 0; kk < DM; kk += 4) {
        const int kb = kk + 2 * kh;
        v2f a, b;
        a.x = o[ml * DM + kb];
        a.y = o[ml * DM + kb + 1];
        b.x = Wo[(size_t)kb * DM + col];
        b.y = Wo[(size_t)(kb + 1) * DM + col];
        c = __builtin_amdgcn_wmma_f32_16x16x4_f32(false, a, false, b,
                                                  (short)0, c, false, false);
    }
    const float bias_c = bo[col];
#pragma unroll
    for (int r = 0; r < 8; ++r)
        out[(r + 8 * kh) * DM + col] = c[r] + bias_c;
}

// ---------------------------------------------------------------------------
extern "C" void kernel_launch(void* const* d_in, const int* in_sizes, int n_in,
                              void* d_out, int out_size, void* d_ws, size_t ws_size,
                              hipStream_t stream) {
    (void)in_sizes; (void)n_in; (void)out_size; (void)ws_size;
    const float* x   = (const float*)d_in[0];
    const float* ck  = (const float*)d_in[1];
    const float* cv  = (const float*)d_in[2];
    const float* Wq  = (const float*)d_in[3];
    const float* bq  = (const float*)d_in[4];
    const float* Wk  = (const float*)d_in[5];
    const float* bk  = (const float*)d_in[6];
    const float* Wv  = (const float*)d_in[7];
    const float* bv  = (const float*)d_in[8];
    const float* Wo  = (const float*)d_in[9];
    const float* bo  = (const float*)d_in[10];
    const int*   ci  = (const int*)d_in[11];
    float* out = (float*)d_out;

    float* ws      = (float*)d_ws;
    float* ws_q    = ws;                         // 16*1024
    float* ws_k    = ws_q + B_ * DM;             // 16*1024
    float* ws_v    = ws_k + B_ * DM;             // 16*1024
    float* cos_tab = ws_v + B_ * DM;             // 4096*32
    float* sin_tab = cos_tab + TMAX * HALF;      // 4096*32
    float* ws_o    = sin_tab + TMAX * HALF;      // 16*1024
    float* ws_part = ws_o + B_ * DM;             // 256*8*66

    qkv_proj_wmma<<<dim3(64, 3), 32, 0, stream>>>(x, Wq, bq, Wk, bk, Wv, bv,
                                                  ws_q, ws_k, ws_v);
    rope_setup<<<(TMAX * HALF + B_ * H_ * HALF) / 256, 256, 0, stream>>>(
        cos_tab, sin_tab, ws_q, ci);
    attn_partial<<<B_ * H_ * NPART, 256, 0, stream>>>(
        ck, cv, ws_q, ws_k, ws_v, cos_tab, sin_tab, ws_part, ci);
    attn_combine<<<B_ * H_, 64, 0, stream>>>(ws_part, ws_o);
    out_proj_wmma<<<64, 32, 0, stream>>>(ws_o, Wo, bo, out);
}